// GEncoder_3959959847494
// MI455X (gfx1250) — compile-verified
//
#include <hip/hip_runtime.h>
#include <math.h>

#define HID 256
#define NGRAPH 512

typedef float v2f __attribute__((ext_vector_type(2)));
typedef float v8f __attribute__((ext_vector_type(8)));

__device__ __forceinline__ float eluf(float x) { return x > 0.f ? x : expm1f(x); }

// ---------------------------------------------------------------------------
// WMMA fp32 GEMM: C[M,N] = act((A[M,K] @ B[K,N] + bias) * preScale)
// act: 0=none 1=relu 2=elu. bd=1: block-diagonal (4 chunks of 64 cols, each
// with its own 64x64 B block; B laid out as [4][64][64], bias as [256]).
// stats!=null: atomically accumulate sum and sum-of-squares of *stored*
// outputs (wave shfl-reduced, 2 atomics per wave) for fused full LayerNorm.
//
// Register blocking: each wave owns a 32x32 C patch = 2 M-tiles x 2 N-tiles
// (four v8f accumulators); per k-step 2 v2f A loads + 4 scalar B loads feed
// 4 x V_WMMA_F32_16X16X4_F32. Addressing uses stable per-lane 64-bit
// pointers advanced by constants so the unrolled k-step deltas fold into the
// 24-bit immediate offset field of global_load (empirically best codegen).
// Block = 256 threads = 8 waves arranged 4(M) x 2(N) -> 128x64 C per block.
// ---------------------------------------------------------------------------
__global__ __launch_bounds__(256) void wmma_gemm_f32(
    const float* __restrict__ A, const float* __restrict__ B,
    const float* __restrict__ bias, float* __restrict__ C,
    int M, int Ktot, int Ntot, int lda, int ldb, int ldc,
    int act, int bd, float preScale, float* __restrict__ stats)
{
    const int wave = threadIdx.x >> 5;
    const int lane = threadIdx.x & 31;
    const int wm = wave & 3;           // M direction within block
    const int wn = wave >> 2;          // N direction within block
    const int m0 = (blockIdx.x * 4 + wm) * 32;
    const int n0 = (blockIdx.y * 2 + wn) * 32;
    if (m0 >= M || n0 >= Ntot) return; // wave-uniform; EXEC stays all-ones

    int kLen = Ktot;
    const float* Bp = B;
    int nB = n0;
    int aColOff = 0;
    if (bd) {
        const int c = n0 >> 6;          // 64-wide chunk (32-col patch stays inside)
        aColOff = c * 64;
        kLen = 64;
        Bp = B + (size_t)c * 64 * ldb;  // ldb == 64 in bd mode
        nB = n0 & 63;
    }

    // A layout (16x4 f32): lanes 0-15 = rows M0..15 holding K={0,1};
    // lanes 16-31 = same rows holding K={2,3}.
    const int rlane = lane & 15;
    const int kk = (lane >> 4) * 2;
    const int nCol = rlane;

    const int mA0 = m0 + rlane;
    int mA1 = m0 + 16 + rlane;
    if (mA1 >= M) mA1 = mA0;            // clamp: ghost rows never stored
    const float* pA0 = A + (size_t)mA0 * lda + aColOff + kk;
    const float* pA1 = A + (size_t)mA1 * lda + aColOff + kk;

    // Two B row pointers (rows kb and kb+1); the +16-column partner rides in
    // the load's immediate offset (+64 bytes).
    const float* pB0 = Bp + (size_t)kk * ldb + nB + nCol;
    const float* pB1 = pB0 + (size_t)ldb;
    const size_t stepB = 4 * (size_t)ldb;

    v8f acc00 = {0.f,0.f,0.f,0.f,0.f,0.f,0.f,0.f};
    v8f acc01 = acc00, acc10 = acc00, acc11 = acc00;

    #pragma unroll 8
    for (int k = 0; k < kLen; k += 4) {
        v2f a0 = *reinterpret_cast<const v2f*>(pA0 + k);
        v2f a1 = *reinterpret_cast<const v2f*>(pA1 + k);
        v2f b0, b1;
        b0.x = pB0[0];   b0.y = pB1[0];
        b1.x = pB0[16];  b1.y = pB1[16];
        pB0 += stepB;
        pB1 += stepB;
        acc00 = __builtin_amdgcn_wmma_f32_16x16x4_f32(false, a0, false, b0, (short)0, acc00, false, false);
        acc01 = __builtin_amdgcn_wmma_f32_16x16x4_f32(false, a0, false, b1, (short)0, acc01, false, false);
        acc10 = __builtin_amdgcn_wmma_f32_16x16x4_f32(false, a1, false, b0, (short)0, acc10, false, false);
        acc11 = __builtin_amdgcn_wmma_f32_16x16x4_f32(false, a1, false, b1, (short)0, acc11, false, false);
    }

    // C/D layout: VGPR r -> row base+r (lanes 0-15) / base+8+r (lanes 16-31),
    // column ntile + (lane&15).
    const int rowHalf = (lane < 16) ? 0 : 8;
    const float bv0 = bias ? bias[n0 + nCol] : 0.f;
    const float bv1 = bias ? bias[n0 + 16 + nCol] : 0.f;
    float s1 = 0.f, s2 = 0.f;

    auto emit = [&](float accv, int m, int col, float bval) {
        float v = (accv + bval) * preScale;
        if (act == 1)      v = fmaxf(v, 0.f);
        else if (act == 2) v = eluf(v);
        if (m < M) {
            C[(size_t)m * ldc + col] = v;
            s1 += v; s2 += v * v;
        }
    };

    #pragma unroll
    for (int r = 0; r < 8; ++r) {
        const int mT0 = m0 + r + rowHalf;
        const int mT1 = m0 + 16 + r + rowHalf;
        emit(acc00[r], mT0, n0 + nCol,      bv0);
        emit(acc01[r], mT0, n0 + 16 + nCol, bv1);
        emit(acc10[r], mT1, n0 + nCol,      bv0);
        emit(acc11[r], mT1, n0 + 16 + nCol, bv1);
    }
    if (stats) {
        for (int o = 16; o > 0; o >>= 1) {
            s1 += __shfl_down(s1, o, 32);
            s2 += __shfl_down(s2, o, 32);
        }
        if (lane == 0) { atomicAdd(&stats[0], s1); atomicAdd(&stats[1], s2); }
    }
}

// --------------------------------------------------------------------------
__global__ void zero_f32(float* p, int n) {
    int i = blockIdx.x * blockDim.x + threadIdx.x;
    if (i < n) p[i] = 0.f;
}
__global__ void zero_i32(int* p, int n) {
    int i = blockIdx.x * blockDim.x + threadIdx.x;
    if (i < n) p[i] = 0;
}

// agg[i] = 2*(g[i] + vn_row(i))   (self term + GIN self loop), float4-wide
__global__ void init_agg_kernel(const float* __restrict__ g,
                                const float* __restrict__ vn,
                                const int* __restrict__ batch, int useBatch,
                                float* __restrict__ agg, int N)
{
    int t = blockIdx.x * blockDim.x + threadIdx.x;
    if (t >= N * 64) return;
    int i = t >> 6, q = t & 63;
    float4 gv = reinterpret_cast<const float4*>(g + (size_t)i * HID)[q];
    const float* vr = useBatch ? vn + (size_t)batch[i] * HID : vn;
    float4 vv = reinterpret_cast<const float4*>(vr)[q];
    float4 o;
    o.x = 2.f * (gv.x + vv.x); o.y = 2.f * (gv.y + vv.y);
    o.z = 2.f * (gv.z + vv.z); o.w = 2.f * (gv.w + vv.w);
    reinterpret_cast<float4*>(agg + (size_t)i * HID)[q] = o;
}

// agg[dst] += g[src] + vn_row(src)   per edge, float4 gather + 4 f32 atomics
__global__ void scatter_edges_kernel(const float* __restrict__ g,
                                     const float* __restrict__ vn,
                                     const int* __restrict__ batch, int useBatch,
                                     const int* __restrict__ src,
                                     const int* __restrict__ dst,
                                     int E, float* __restrict__ agg)
{
    int t = blockIdx.x * blockDim.x + threadIdx.x;
    if (t >= E * 64) return;
    int e = t >> 6, q = t & 63;
    int s = src[e], d = dst[e];
    float4 gv = reinterpret_cast<const float4*>(g + (size_t)s * HID)[q];
    const float* vr = useBatch ? vn + (size_t)batch[s] * HID : vn;
    float4 vv = reinterpret_cast<const float4*>(vr)[q];
    float* out = agg + (size_t)d * HID + q * 4;
    atomicAdd(out + 0, gv.x + vv.x);
    atomicAdd(out + 1, gv.y + vv.y);
    atomicAdd(out + 2, gv.z + vv.z);
    atomicAdd(out + 3, gv.w + vv.w);
}

__global__ void bcast_rows(const float* __restrict__ row, float* __restrict__ out,
                           int rows, int cols)
{
    int t = blockIdx.x * blockDim.x + threadIdx.x;
    if (t < rows * cols) out[t] = row[t % cols];
}

// vt[batch[i]] += g[i]
__global__ void segsum_nodes(const float* __restrict__ g, const int* __restrict__ batch,
                             float* __restrict__ vt, int N)
{
    int t = blockIdx.x * blockDim.x + threadIdx.x;
    if (t >= N * 64) return;
    int i = t >> 6, q = t & 63;
    float4 gv = reinterpret_cast<const float4*>(g + (size_t)i * HID)[q];
    float* out = vt + (size_t)batch[i] * HID + q * 4;
    atomicAdd(out + 0, gv.x); atomicAdd(out + 1, gv.y);
    atomicAdd(out + 2, gv.z); atomicAdd(out + 3, gv.w);
}

// out = (h - mean) * rsqrt(var + eps), mean/var from global sum/sumsq
__global__ void ln_apply_kernel(const float* __restrict__ h,
                                const float* __restrict__ stats,
                                float invCount, float* __restrict__ out, int total4)
{
    int t = blockIdx.x * blockDim.x + threadIdx.x;
    if (t >= total4) return;
    float mean = stats[0] * invCount;
    float var = stats[1] * invCount - mean * mean;
    float r = rsqrtf(var + 1e-5f);
    float4 v = reinterpret_cast<const float4*>(h)[t];
    float4 o;
    o.x = (v.x - mean) * r; o.y = (v.y - mean) * r;
    o.z = (v.z - mean) * r; o.w = (v.w - mean) * r;
    reinterpret_cast<float4*>(out)[t] = o;
}

__global__ void count_nodes_kernel(const int* __restrict__ batch, int N, int* __restrict__ cnt)
{
    int i = blockIdx.x * blockDim.x + threadIdx.x;
    if (i < N) atomicAdd(&cnt[batch[i]], 1);
}

// exclusive scan of 512 counts, single block
__global__ void scan512(const int* __restrict__ cnt, int* __restrict__ off)
{
    __shared__ int s[NGRAPH];
    int t = threadIdx.x;
    int v0 = cnt[t];
    s[t] = v0;
    __syncthreads();
    for (int d = 1; d < NGRAPH; d <<= 1) {
        int v = (t >= d) ? s[t - d] : 0;
        __syncthreads();
        s[t] += v;
        __syncthreads();
    }
    off[t] = s[t] - v0;
}

// per-graph max/mean pooling over concat([g0,g1,g2]); batch is sorted so each
// graph owns a contiguous node range. One block per graph, thread t owns
// feature t of each of the three blocks.
__global__ __launch_bounds__(256) void readout_kernel(
    const float* __restrict__ g0, const float* __restrict__ g1,
    const float* __restrict__ g2, const int* __restrict__ off,
    const int* __restrict__ cnt, float* __restrict__ readb)
{
    int g = blockIdx.x, t = threadIdx.x;
    int s = off[g], c = cnt[g];
    float mx0 = -INFINITY, mx1 = -INFINITY, mx2 = -INFINITY;
    float sm0 = 0.f, sm1 = 0.f, sm2 = 0.f;
    for (int i = s; i < s + c; ++i) {
        size_t base = (size_t)i * HID + t;
        float v0 = g0[base], v1 = g1[base], v2 = g2[base];
        mx0 = fmaxf(mx0, v0); sm0 += v0;
        mx1 = fmaxf(mx1, v1); sm1 += v1;
        mx2 = fmaxf(mx2, v2); sm2 += v2;
    }
    float inv = 1.f / (float)(c > 1 ? c : 1);
    float* rg = readb + (size_t)g * 1536;
    rg[t] = mx0; rg[256 + t] = mx1; rg[512 + t] = mx2;
    rg[768 + t] = sm0 * inv; rg[1024 + t] = sm1 * inv; rg[1280 + t] = sm2 * inv;
}

// ssl[g,k] = 0.05 + 0.35*sigmoid(dot(x5[g, k*64:(k+1)*64], dhW[k]) + dhb[k])
__global__ void ssl_kernel(const float* __restrict__ x5, const float* __restrict__ dhW,
                           const float* __restrict__ dhb, float* __restrict__ out)
{
    int idx = blockIdx.x * blockDim.x + threadIdx.x;
    if (idx >= NGRAPH * 3) return;
    int g = idx / 3, k = idx % 3;
    const float* xv = x5 + (size_t)g * HID + k * 64;
    const float* w = dhW + k * 64;
    float z = dhb[k];
    for (int j = 0; j < 64; ++j) z += xv[j] * w[j];
    out[idx] = 0.05f + 0.35f / (1.f + expf(-z));
}

// logits = log_softmax(x5 @ clsW + clsb) over the 2 classes
__global__ void logits_kernel(const float* __restrict__ x5, const float* __restrict__ clsW,
                              const float* __restrict__ clsb, float* __restrict__ out)
{
    int g = blockIdx.x * blockDim.x + threadIdx.x;
    if (g >= NGRAPH) return;
    const float* xv = x5 + (size_t)g * HID;
    float l0 = clsb[0], l1 = clsb[1];
    for (int j = 0; j < HID; ++j) {
        float v = xv[j];
        l0 += v * clsW[j * 2];
        l1 += v * clsW[j * 2 + 1];
    }
    float m = fmaxf(l0, l1);
    float lse = m + logf(expf(l0 - m) + expf(l1 - m));
    out[g * 2] = l0 - lse;
    out[g * 2 + 1] = l1 - lse;
}

// ---------------------------------------------------------------------------
extern "C" void kernel_launch(void* const* d_in, const int* in_sizes, int n_in,
                              void* d_out, int out_size, void* d_ws, size_t ws_size,
                              hipStream_t stream)
{
    (void)n_in; (void)out_size; (void)ws_size;
    const float* x      = (const float*)d_in[0];
    const int*   ei     = (const int*)d_in[1];
    const int*   batch  = (const int*)d_in[2];
    const float* lin1_W = (const float*)d_in[3];
    const float* lin1_b = (const float*)d_in[4];
    const float* g0W1   = (const float*)d_in[5];
    const float* g0b1   = (const float*)d_in[6];
    const float* g0W2   = (const float*)d_in[7];
    const float* g0b2   = (const float*)d_in[8];
    const float* g1W1   = (const float*)d_in[9];
    const float* g1b1   = (const float*)d_in[10];
    const float* g1W2   = (const float*)d_in[11];
    const float* g1b2   = (const float*)d_in[12];
    const float* vn_emb = (const float*)d_in[13];
    const float* vnW1   = (const float*)d_in[14];
    const float* vnb1   = (const float*)d_in[15];
    const float* vnW2   = (const float*)d_in[16];
    const float* vnb2   = (const float*)d_in[17];
    const float* loW    = (const float*)d_in[18];
    const float* lob    = (const float*)d_in[19];
    const float* clsW   = (const float*)d_in[20];
    const float* clsb   = (const float*)d_in[21];
    const float* dhW    = (const float*)d_in[22];
    const float* dhb    = (const float*)d_in[23];

    const int N = in_sizes[0] / 128;    // 50000
    const int E = in_sizes[1] / 2;      // 800000
    const int* src = ei;
    const int* dst = ei + E;

    // workspace layout
    float* ws = (float*)d_ws;
    const size_t NF = (size_t)N * HID;
    float* g0    = ws; ws += NF;
    float* g1    = ws; ws += NF;
    float* g2    = ws; ws += NF;
    float* agg   = ws; ws += NF;
    float* tmp   = ws; ws += NF;
    float* vt    = ws; ws += (size_t)NGRAPH * HID;
    float* vnh   = ws; ws += (size_t)NGRAPH * 512;
    float* vn1   = ws; ws += (size_t)NGRAPH * HID;
    float* readb = ws; ws += (size_t)NGRAPH * 1536;
    float* stats = ws; ws += 4;
    int* cnt = (int*)ws; ws += NGRAPH;
    int* off = (int*)ws;

    const float BN_S = 1.0f / sqrtf(1.0f + 1e-5f);
    const float invCnt = 1.0f / (float)((size_t)N * HID);
    const int tpb = 256;
    const int gNQ = (N * 64 + tpb - 1) / tpb;
    const int gEQ = (E * 64 + tpb - 1) / tpb;

    auto gemm = [&](const float* A, const float* B, const float* bias, float* C,
                    int M, int K, int Nn, int lda, int ldb, int ldc,
                    int act, int bd, float preScale, float* st) {
        dim3 grid((M + 127) / 128, Nn / 64);   // 128 rows x 64 cols per block
        wmma_gemm_f32<<<grid, dim3(256), 0, stream>>>(
            A, B, bias, C, M, K, Nn, lda, ldb, ldc, act, bd, preScale, st);
    };

    // 1) h = elu(x @ lin1_W + b)  -> g0  (gr[0])
    gemm(x, lin1_W, lin1_b, g0, N, 128, 256, 128, 256, 256, /*elu*/2, 0, 1.f, nullptr);

    // 2) layer 0 (full-width GIN). vn = broadcast(vn_emb) at this point.
    init_agg_kernel<<<gNQ, tpb, 0, stream>>>(g0, vn_emb, nullptr, 0, agg, N);
    scatter_edges_kernel<<<gEQ, tpb, 0, stream>>>(g0, vn_emb, nullptr, 0, src, dst, E, agg);
    gemm(agg, g0W1, g0b1, tmp, N, 256, 256, 256, 256, 256, /*relu*/1, 0, 1.f, nullptr);
    zero_f32<<<1, 32, 0, stream>>>(stats, 4);
    gemm(tmp, g0W2, g0b2, agg, N, 256, 256, 256, 256, 256, /*elu*/2, 0, 1.f, stats);
    ln_apply_kernel<<<gNQ, tpb, 0, stream>>>(agg, stats, invCnt, g1, N * 64);

    // virtual node update: vt = segsum(g0) + vn_emb, then 2x relu(bn(linear))
    bcast_rows<<<(NGRAPH * HID + tpb - 1) / tpb, tpb, 0, stream>>>(vn_emb, vt, NGRAPH, HID);
    segsum_nodes<<<gNQ, tpb, 0, stream>>>(g0, batch, vt, N);
    gemm(vt,  vnW1, vnb1, vnh, NGRAPH, 256, 512, 256, 512, 512, /*relu*/1, 0, BN_S, nullptr);
    gemm(vnh, vnW2, vnb2, vn1, NGRAPH, 512, 256, 512, 256, 256, /*relu*/1, 0, BN_S, nullptr);

    // 3) layer 1 (block-diagonal GIN on 4 chunks of 64)
    init_agg_kernel<<<gNQ, tpb, 0, stream>>>(g1, vn1, batch, 1, agg, N);
    scatter_edges_kernel<<<gEQ, tpb, 0, stream>>>(g1, vn1, batch, 1, src, dst, E, agg);
    gemm(agg, g1W1, g1b1, tmp, N, 256, 256, 256, 64, 256, /*relu*/1, /*bd*/1, 1.f, nullptr);
    zero_f32<<<1, 32, 0, stream>>>(stats, 4);
    gemm(tmp, g1W2, g1b2, agg, N, 256, 256, 256, 64, 256, /*elu*/2, /*bd*/1, 1.f, stats);
    ln_apply_kernel<<<gNQ, tpb, 0, stream>>>(agg, stats, invCnt, g2, N * 64);

    // 4) readout: segment max + mean over concat([g0,g1,g2])
    zero_i32<<<(NGRAPH + tpb - 1) / tpb, tpb, 0, stream>>>(cnt, NGRAPH);
    count_nodes_kernel<<<(N + tpb - 1) / tpb, tpb, 0, stream>>>(batch, N, cnt);
    scan512<<<1, NGRAPH, 0, stream>>>(cnt, off);
    readout_kernel<<<NGRAPH, 256, 0, stream>>>(g0, g1, g2, off, cnt, readb);

    // 5) x5 = elu(read @ lin_out_W + b)  -> written straight into d_out
    float* outf = (float*)d_out;
    float* x5 = outf + NGRAPH * 2;
    gemm(readb, loW, lob, x5, NGRAPH, 1536, 256, 1536, 256, 256, /*elu*/2, 0, 1.f, nullptr);

    // 6) ssl head and 7) classifier log-softmax
    ssl_kernel<<<(NGRAPH * 3 + tpb - 1) / tpb, tpb, 0, stream>>>(
        x5, dhW, dhb, outf + NGRAPH * 2 + (size_t)NGRAPH * HID);
    logits_kernel<<<(NGRAPH + tpb - 1) / tpb, tpb, 0, stream>>>(x5, clsW, clsb, outf);
}